// OpMessagePassing_59768764891825
// MI455X (gfx1250) — compile-verified
//
#include <hip/hip_runtime.h>
#include <stdint.h>

// ---------------------------------------------------------------------------
// spspmm message passing:  out[a] += Aval[c] * Bval[d]   (rows of length D=32)
//
// Memory-bound (AI ~0.16 flop/byte; HBM roofline ~70us at 23.3 TB/s).
// CDNA5 paths used:
//   * global_load_async_to_lds_b32 + s_wait_asynccnt : 4-deep per-wave
//     pipelined row gathers into LDS (ASYNCcnt-tracked, in-order completion)
//   * explicit ds_load_b32 + s_wait_dscnt consume path
//   * global_prefetch_b8 on the sequential index streams
//   * hardware global_atomic_add_f32 (no-return) for the segment scatter-sum
//
// Wave32 layout: lane == feature (D=32). One pair per wave per pipeline stage.
// All index/loop arithmetic is 32-bit (indices < 2^21, offsets < 2^28).
// ---------------------------------------------------------------------------

#define DFEAT            32
#define WAVES_PER_BLOCK  8
#define BLOCK            (WAVES_PER_BLOCK * 32)
#define GRID             4096   // 32768 waves -> ~122 pairs/wave for M=4e6
#define NBUF             4      // pipeline depth (async stages in flight)

#if __has_builtin(__builtin_amdgcn_s_wait_asynccnt)
#define WAIT_ASYNC(n) __builtin_amdgcn_s_wait_asynccnt(n)
#else
#define WAIT_ASYNC(n) asm volatile("s_wait_asynccnt %0" ::"n"(n) : "memory")
#endif

// One async b32 load: all 32 lanes each move 4B (one 128B row per instruction,
// one ASYNCcnt increment). GV addressing: 64-bit per-lane address in a VGPR
// pair, LDS byte address in VDST.
__device__ __forceinline__ void async_row_to_lds(uint32_t lds_byte_addr,
                                                 unsigned long long gaddr) {
    asm volatile("global_load_async_to_lds_b32 %0, %1, off"
                 :
                 : "v"(lds_byte_addr), "v"(gaddr)
                 : "memory");
}

// Consume one staged pair from LDS on the DS path: A at +0, B at +128.
// The trailing s_wait_dscnt makes the values ready AND makes it safe to
// immediately refill this ring slot with a new async load.
__device__ __forceinline__ void lds_consume_pair(uint32_t lds_byte_addr,
                                                 float& a, float& b) {
    asm volatile("ds_load_b32 %0, %2\n\t"
                 "ds_load_b32 %1, %2 offset:128\n\t"
                 "s_wait_dscnt 0x0"
                 : "=v"(a), "=v"(b)
                 : "v"(lds_byte_addr)
                 : "memory");
}

__global__ __launch_bounds__(BLOCK) void spspmm_gather_scatter_kernel(
    const float* __restrict__ Aval,
    const float* __restrict__ Bval,
    const unsigned* __restrict__ idxA,  // low dwords of int64 a[], stride 2
    const unsigned* __restrict__ idxC,  // low dwords of int64 c[], stride 2
    const unsigned* __restrict__ idxD,  // low dwords of int64 d[], stride 2
    float* __restrict__ out,            // [tar_nnz][32], pre-zeroed
    int M)
{
    // Per-wave 4-deep ring: [wave][buf][A|B][lane] -> 256B per buffer stage.
    __shared__ float sbuf[WAVES_PER_BLOCK][NBUF][2][DFEAT];

    const int lane  = threadIdx.x & 31;
    const int lwave = threadIdx.x >> 5;
    const int wid   = (int)blockIdx.x * WAVES_PER_BLOCK + lwave;
    const int nw    = (int)gridDim.x * WAVES_PER_BLOCK;

    // Low 32 bits of a generic LDS pointer == LDS byte offset for DS/async ops.
    const uint32_t lds0 = (uint32_t)(uintptr_t)&sbuf[lwave][0][0][lane];
    const unsigned long long baseA =
        (unsigned long long)(uintptr_t)Aval + (unsigned)(lane * 4);
    const unsigned long long baseB =
        (unsigned long long)(uintptr_t)Bval + (unsigned)(lane * 4);

    int p = wid;
    if (p >= M) return;

    // ---- prologue: issue up to NBUF stages (2 async loads per stage) ----
    {
        int q = p;
        for (int j = 0; j < NBUF && q < M; ++j, q += nw) {
            const unsigned c = idxC[2 * q];
            const unsigned d = idxD[2 * q];
            const uint32_t l = lds0 + (uint32_t)j * 256u;
            async_row_to_lds(l,        baseA + (unsigned long long)c * (DFEAT * 4));
            async_row_to_lds(l + 128u, baseB + (unsigned long long)d * (DFEAT * 4));
        }
    }

    int pi = p + NBUF * nw;  // next stage to issue
    uint32_t buf = 0;
    while (p < M) {
        // Stage p is the oldest in flight; async loads complete in order, so
        // wait until at most 2*(stages issued beyond p) remain outstanding.
        if (p + 3 * nw < M)      WAIT_ASYNC(6);
        else if (p + 2 * nw < M) WAIT_ASYNC(4);
        else if (p + nw < M)     WAIT_ASYNC(2);
        else                     WAIT_ASYNC(0);

        // Prefetch the sequential index streams well ahead
        // (-> global_prefetch_b8; speculative, no counter traffic).
        {
            const int pf = p + 8 * nw;
            if (pf < M) {
                __builtin_prefetch(idxA + 2 * pf, 0, 0);
                __builtin_prefetch(idxC + 2 * pf, 0, 0);
                __builtin_prefetch(idxD + 2 * pf, 0, 0);
            }
        }

        // Consume current stage from LDS (ds_load_b32 x2 + s_wait_dscnt).
        float av, bv;
        lds_consume_pair(lds0 + buf * 256u, av, bv);

        // Scatter-add into the target row: 32 lanes -> one 128B line.
        const unsigned a = idxA[2 * p];
        unsafeAtomicAdd(&out[a * DFEAT + (unsigned)lane], av * bv);

        // Refill the ring slot we just drained (s_wait_dscnt above guarantees
        // the ds reads of this buffer completed before the async overwrite).
        if (pi < M) {
            const unsigned c = idxC[2 * pi];
            const unsigned d = idxD[2 * pi];
            const uint32_t l = lds0 + buf * 256u;
            async_row_to_lds(l,        baseA + (unsigned long long)c * (DFEAT * 4));
            async_row_to_lds(l + 128u, baseB + (unsigned long long)d * (DFEAT * 4));
        }

        pi += nw;
        p += nw;
        buf = (buf + 1u) & (NBUF - 1u);
    }
}

extern "C" void kernel_launch(void* const* d_in, const int* in_sizes, int n_in,
                              void* d_out, int out_size, void* d_ws, size_t ws_size,
                              hipStream_t stream) {
    const float* Aval = (const float*)d_in[0];
    const float* Bval = (const float*)d_in[1];
    // acd is int64 [3][M]; all values < 2^21, so read the low dwords
    // (little-endian) as unsigned with stride 2.
    const unsigned* idx32 = (const unsigned*)d_in[2];
    // d_in[3] = tar_nnz scalar; out_size == tar_nnz * 32 already.

    const int M = in_sizes[2] / 3;
    const unsigned* idxA = idx32;                    // a: target nnz index
    const unsigned* idxC = idx32 + 2 * (size_t)M;    // c: A row index
    const unsigned* idxD = idx32 + 4 * (size_t)M;    // d: B row index
    float* out = (float*)d_out;

    // Zero the accumulation target (harness poisons d_out; atomics accumulate).
    hipMemsetAsync(d_out, 0, (size_t)out_size * sizeof(float), stream);

    spspmm_gather_scatter_kernel<<<GRID, BLOCK, 0, stream>>>(
        Aval, Bval, idxA, idxC, idxD, out, M);
}